// MyModule11_45792941310687
// MI455X (gfx1250) — compile-verified
//
#include <hip/hip_runtime.h>

// Fused 7-layer MLP + dropout masks + softmax for MI455X (gfx1250).
// One wave32 processes a 16-row tile; all dense layers run on the matrix pipe
// via V_WMMA_F32_16X16X4_F32 (3 K-chunks of 4, K/N zero-padded in LDS).
// Memory-bound: ~188 MB total traffic (X + 3 masks + out) ~= 8 us at 23.3 TB/s.
// Branch-free inner pipeline: unconditional clamped global loads, branch-free
// bias+relu epilogue (padded-zero columns are stored, never read back).

typedef float v2f __attribute__((ext_vector_type(2)));
typedef float v8f __attribute__((ext_vector_type(8)));

static __device__ __forceinline__ v8f wmma_f32x4(v2f a, v2f b, v8f c) {
  // (neg_a, A, neg_b, B, c_mod, C, reuse_a, reuse_b)
  return __builtin_amdgcn_wmma_f32_16x16x4_f32(false, a, false, b, (short)0, c,
                                               false, false);
}

#define WAVES_PER_BLOCK 8
#define BLOCK_THREADS 256
#define ASTRIDE 20  // act row pitch (floats): even (8B-aligned pairs) and
                    // bank-conflict-free for both C-layout stores and A-reads

__global__ __launch_bounds__(BLOCK_THREADS) void mlp7_fused_kernel(
    const float* __restrict__ X,
    const float* __restrict__ W1, const float* __restrict__ b1,
    const float* __restrict__ W2, const float* __restrict__ b2,
    const float* __restrict__ W3, const float* __restrict__ b3,
    const float* __restrict__ W4, const float* __restrict__ b4,
    const float* __restrict__ W5, const float* __restrict__ b5,
    const float* __restrict__ W6, const float* __restrict__ b6,
    const float* __restrict__ W7, const float* __restrict__ b7,
    const float* __restrict__ mask1, const float* __restrict__ mask2,
    const float* __restrict__ mask3,
    float* __restrict__ out, int nrows)
{
  // Padded weights: per layer [16 (n, zero rows 12..15)][12 (k, zero-padded)].
  __shared__ float sW[6 * 192];
  __shared__ float sB[6 * 16];
  __shared__ float sW7[24];
  __shared__ float sB7[2];
  __shared__ float sAct[WAVES_PER_BLOCK][16 * ASTRIDE];  // 16 rows x 20 pitch

  const int t = threadIdx.x;

  // ---- Stage weights/biases into LDS (before any wave may exit) ----
  if (t < 192) {
    const int n = t / 12, k = t % 12;
    sW[0 * 192 + t] = (n < 12 && k < 9) ? W1[n * 9 + k] : 0.0f;
    sW[1 * 192 + t] = (n < 12) ? W2[n * 12 + k] : 0.0f;
    sW[2 * 192 + t] = (n < 12) ? W3[n * 12 + k] : 0.0f;
    sW[3 * 192 + t] = (n < 12) ? W4[n * 12 + k] : 0.0f;
    sW[4 * 192 + t] = (n < 12) ? W5[n * 12 + k] : 0.0f;
    sW[5 * 192 + t] = (n < 12) ? W6[n * 12 + k] : 0.0f;
  }
  if (t < 16) {
    sB[0 * 16 + t] = (t < 12) ? b1[t] : 0.0f;
    sB[1 * 16 + t] = (t < 12) ? b2[t] : 0.0f;
    sB[2 * 16 + t] = (t < 12) ? b3[t] : 0.0f;
    sB[3 * 16 + t] = (t < 12) ? b4[t] : 0.0f;
    sB[4 * 16 + t] = (t < 12) ? b5[t] : 0.0f;
    sB[5 * 16 + t] = (t < 12) ? b6[t] : 0.0f;
  }
  if (t < 24) sW7[t] = W7[t];
  if (t < 2)  sB7[t] = b7[t];
  __syncthreads();

  const int wave = t >> 5;        // wave32
  const int lane = t & 31;
  const int lh   = lane >> 4;     // half: 0 = lanes 0-15, 1 = lanes 16-31
  const int lrow = lane & 15;     // A-layout row / B-&C-layout column (n)
  const int tile = blockIdx.x * WAVES_PER_BLOCK + wave;
  const int nTiles = (nrows + 15) >> 4;
  if (tile >= nTiles) return;     // whole-wave exit -> EXEC stays all-1s

  const long base_row = (long)tile << 4;
  float* __restrict__ act = sAct[wave];

  // Clamped row index: every global load is in-bounds without branching.
  const long grow = min(base_row + (long)lrow, (long)nrows - 1);

  // Per-lane bias registers (lane's C column is fixed at n = lrow).
  float biasr[6];
#pragma unroll
  for (int l = 0; l < 6; ++l) biasr[l] = sB[l * 16 + lrow];

  // ---- Prefetch all dropout-mask fragments into registers (9x b64 loads) ----
  v2f m1v[3], m2v[3], m3v[3];
  {
    const float* r1 = mask1 + grow * 12;
    const float* r2 = mask2 + grow * 12;
    const float* r3 = mask3 + grow * 12;
#pragma unroll
    for (int ch = 0; ch < 3; ++ch) {
      const int k0 = ch * 4 + lh * 2;          // 8B-aligned pairs
      m1v[ch] = *(const v2f*)(r1 + k0);
      m2v[ch] = *(const v2f*)(r2 + k0);
      m3v[ch] = *(const v2f*)(r3 + k0);
    }
  }

  // ---- Stage X tile (16x9, coalesced 576B) into act, zero-pad k=9..11 ----
  {
    const long xbase = base_row * 9;
    const long xmax  = (long)nrows * 9 - 1;
#pragma unroll
    for (int it = 0; it < 5; ++it) {
      const int i = lane + it * 32;
      if (i < 144) {
        const float v = X[min(xbase + i, xmax)];
        act[(i / 9) * ASTRIDE + (i % 9)] = v;
      }
    }
#pragma unroll
    for (int it = 0; it < 2; ++it) {
      const int i = lane + it * 32;
      if (i < 48) act[(i / 3) * ASTRIDE + 9 + (i % 3)] = 0.0f;  // kill stale NaNs
    }
  }

  // C layout -> LDS, fusing bias + relu. Lane holds column n=lrow, rows j+8*lh.
  // Branch-free: columns 12..15 are exact zeros (B rows padded to 0), stored
  // but never read back (A-fragments only touch k < 12).
  auto epilogue = [&](const v8f cc, int li) {
    const float bias = biasr[li];
#pragma unroll
    for (int j = 0; j < 8; ++j) {
      float v = cc[j] + bias;
      act[(j + 8 * lh) * ASTRIDE + lrow] = v > 0.0f ? v : 0.0f;
    }
  };

  // One dense layer: A-fragments from act (optionally masked from registers),
  // B-fragments from padded LDS weights. 3x wmma_f32_16x16x4.
  auto layer = [&](int li, const v2f* mv) -> v8f {
    v8f cc = {};
#pragma unroll
    for (int ch = 0; ch < 3; ++ch) {
      const int k0 = ch * 4 + lh * 2;
      v2f a = *(const v2f*)&act[lrow * ASTRIDE + k0];  // 8B-aligned
      if (mv) {
        a.x *= mv[ch].x;
        a.y *= mv[ch].y;
      }
      const v2f bb = *(const v2f*)&sW[li * 192 + lrow * 12 + k0];
      cc = wmma_f32x4(a, bb, cc);
    }
    return cc;
  };

  v8f c;
  c = layer(0, nullptr); epilogue(c, 0);   // h1 = relu(X W1' + b1)
  c = layer(1, nullptr); epilogue(c, 1);   // h2 = relu(h1 W2' + b2)
  c = layer(2, m1v);     epilogue(c, 2);   // h3 = relu((h2*mask1) W3' + b3)
  c = layer(3, nullptr); epilogue(c, 3);   // h4 = relu(h3 W4' + b4)
  c = layer(4, m2v);     epilogue(c, 4);   // h5 = relu((h4*mask2) W5' + b5)
  c = layer(5, m3v);     epilogue(c, 5);   // h6 = relu((h5*mask3) W6' + b6)

  // ---- Layer 7 (12 -> 2) + softmax, scalar on lanes 0-15 (one row each) ----
  if (lane < 16 && (base_row + lane) < (long)nrows) {
    const float* hr = act + lane * ASTRIDE;
    float z0 = sB7[0], z1 = sB7[1];
#pragma unroll
    for (int k = 0; k < 12; ++k) {
      const float h = hr[k];
      z0 = fmaf(h, sW7[k], z0);
      z1 = fmaf(h, sW7[12 + k], z1);
    }
    const float mx = fmaxf(z0, z1);
    const float e0 = __expf(z0 - mx);
    const float e1 = __expf(z1 - mx);
    const float inv = 1.0f / (e0 + e1);
    float2 o;
    o.x = e0 * inv;
    o.y = e1 * inv;
    *(float2*)(out + (base_row + lane) * 2) = o;  // coalesced 8B store
  }
}

extern "C" void kernel_launch(void* const* d_in, const int* in_sizes, int n_in,
                              void* d_out, int out_size, void* d_ws, size_t ws_size,
                              hipStream_t stream) {
  const float* X  = (const float*)d_in[0];
  const float* W1 = (const float*)d_in[1];  const float* b1 = (const float*)d_in[2];
  const float* W2 = (const float*)d_in[3];  const float* b2 = (const float*)d_in[4];
  const float* W3 = (const float*)d_in[5];  const float* b3 = (const float*)d_in[6];
  const float* W4 = (const float*)d_in[7];  const float* b4 = (const float*)d_in[8];
  const float* W5 = (const float*)d_in[9];  const float* b5 = (const float*)d_in[10];
  const float* W6 = (const float*)d_in[11]; const float* b6 = (const float*)d_in[12];
  const float* W7 = (const float*)d_in[13]; const float* b7 = (const float*)d_in[14];
  const float* m1 = (const float*)d_in[15];
  const float* m2 = (const float*)d_in[16];
  const float* m3 = (const float*)d_in[17];
  float* out = (float*)d_out;

  const int nrows  = in_sizes[0] / 9;
  const int nTiles = (nrows + 15) / 16;
  const int blocks = (nTiles + WAVES_PER_BLOCK - 1) / WAVES_PER_BLOCK;

  hipLaunchKernelGGL(mlp7_fused_kernel, dim3(blocks), dim3(BLOCK_THREADS), 0,
                     stream, X, W1, b1, W2, b2, W3, b3, W4, b4, W5, b5, W6, b6,
                     W7, b7, m1, m2, m3, out, nrows);
}